// ensemble_linear_80161269612843
// MI455X (gfx1250) — compile-verified
//
#include <hip/hip_runtime.h>

// Ensemble of 32 independent 64x64 linear layers applied to 8192x(32*64) fp32 input.
// y[b, d*64+o] = sum_i x[b, d*64+i] * W[d*64+o, d*64+i]
//
// fp32 path -> V_WMMA_F32_16X16X4_F32 (D = A(16x4) * B(4x16) + C), wave32.

typedef __attribute__((ext_vector_type(2))) float v2f;
typedef __attribute__((ext_vector_type(4))) float v4f;
typedef __attribute__((ext_vector_type(8))) float v8f;

#define BATCH     8192
#define DDET      32
#define IN_S      64
#define OUT_S     64
#define ROWSTRIDE (DDET * IN_S)   // 2048 floats per x/y row
#define LDS_PITCH 66              // 64 + 2 padding: kills bank conflicts on column reads
#define ROWS_PER_BLOCK 128        // 8 waves * 16 rows

__global__ __launch_bounds__(256)
void ensemble_linear_wmma_f32(const float* __restrict__ x,
                              const float* __restrict__ W,
                              float* __restrict__ y)
{
    __shared__ float wl[OUT_S * LDS_PITCH];   // 64x64 weight block, padded (16.5 KB)

    const int d     = blockIdx.x;             // detector index
    const int mblk  = blockIdx.y;             // 128-row batch chunk
    const int tid   = threadIdx.x;
    const int wave  = tid >> 5;
    const int lane  = tid & 31;
    const int lrow  = lane & 15;              // M (for A) / N (for B,C,D) position
    const int khalf = lane >> 4;              // 0: K={0,1}, 1: K={2,3} within a K4 step

    // ---- Stage this detector's 64x64 diagonal block into LDS (float4 loads) ----
    // blocks[d][o][i] = W[(d*64+o)*2048 + d*64 + i]
    {
        const float* wsrc = W + (size_t)(d * OUT_S) * ROWSTRIDE + d * IN_S;
        #pragma unroll
        for (int t = 0; t < 4; ++t) {
            int idx = (tid + t * 256) * 4;    // 4096 floats total
            int o = idx / IN_S;
            int i = idx % IN_S;
            v4f v = *(const v4f*)(wsrc + (size_t)o * ROWSTRIDE + i);
            *(v4f*)(&wl[o * LDS_PITCH + i]) = v;
        }
    }
    __syncthreads();

    // ---- Preload A fragments: 16 rows x 64 K, fp32 16x16x4 A layout ----
    // lane holds row (lrow), K pair {kk*4 + 2*khalf, +1}
    const int rowBase = mblk * ROWS_PER_BLOCK + wave * 16;
    const float* xrow = x + (size_t)(rowBase + lrow) * ROWSTRIDE + d * IN_S;

    v2f afrag[16];
    #pragma unroll
    for (int kk = 0; kk < 16; ++kk)
        afrag[kk] = *(const v2f*)(xrow + kk * 4 + 2 * khalf);

    // ---- Accumulate: 4 N-tiles of 16, 16 K-steps of 4 -> 64 WMMAs ----
    v8f acc0 = {}, acc1 = {}, acc2 = {}, acc3 = {};
    #pragma unroll
    for (int kk = 0; kk < 16; ++kk) {
        const int k0 = kk * 4 + 2 * khalf;
        // B fragment: b[k][n] = blocks[d][n][k]; lane supplies N=lrow(+16*nt), K pair at k0
        v2f b0 = *(const v2f*)(&wl[(0 * 16 + lrow) * LDS_PITCH + k0]);
        v2f b1 = *(const v2f*)(&wl[(1 * 16 + lrow) * LDS_PITCH + k0]);
        v2f b2 = *(const v2f*)(&wl[(2 * 16 + lrow) * LDS_PITCH + k0]);
        v2f b3 = *(const v2f*)(&wl[(3 * 16 + lrow) * LDS_PITCH + k0]);

        acc0 = __builtin_amdgcn_wmma_f32_16x16x4_f32(false, afrag[kk], false, b0,
                                                     (short)0, acc0, false, false);
        acc1 = __builtin_amdgcn_wmma_f32_16x16x4_f32(false, afrag[kk], false, b1,
                                                     (short)0, acc1, false, false);
        acc2 = __builtin_amdgcn_wmma_f32_16x16x4_f32(false, afrag[kk], false, b2,
                                                     (short)0, acc2, false, false);
        acc3 = __builtin_amdgcn_wmma_f32_16x16x4_f32(false, afrag[kk], false, b3,
                                                     (short)0, acc3, false, false);
    }

    // ---- Store: C/D layout: lane(0-15)->N, VGPR v -> M=v; lanes 16-31 -> M=8+v ----
    // For fixed (nt, v): lanes 0-15 write 16 consecutive floats of row rowBase+v,
    // lanes 16-31 the same columns of row rowBase+8+v -> coalesced b32 stores.
    float* yrow = y + (size_t)(rowBase + 8 * khalf) * ROWSTRIDE + d * OUT_S + lrow;
    #pragma unroll
    for (int v = 0; v < 8; ++v) {
        yrow[(size_t)v * ROWSTRIDE + 0 * 16] = acc0[v];
        yrow[(size_t)v * ROWSTRIDE + 1 * 16] = acc1[v];
        yrow[(size_t)v * ROWSTRIDE + 2 * 16] = acc2[v];
        yrow[(size_t)v * ROWSTRIDE + 3 * 16] = acc3[v];
    }
}

extern "C" void kernel_launch(void* const* d_in, const int* in_sizes, int n_in,
                              void* d_out, int out_size, void* d_ws, size_t ws_size,
                              hipStream_t stream)
{
    const float* x = (const float*)d_in[0];
    const float* W = (const float*)d_in[1];
    // d_in[2] = num_detectors (constant 32, baked in)
    float* y = (float*)d_out;

    dim3 grid(DDET, BATCH / ROWS_PER_BLOCK);   // (32, 64)
    ensemble_linear_wmma_f32<<<grid, 256, 0, stream>>>(x, W, y);
}